// VectorQuantization_86311662780510
// MI455X (gfx1250) — compile-verified
//
#include <hip/hip_runtime.h>

typedef __attribute__((ext_vector_type(2))) float v2f;
typedef __attribute__((ext_vector_type(4))) float v4f;
typedef __attribute__((ext_vector_type(8))) float v8f;

static constexpr int B_ = 1024;
static constexpr int N_ = 8192;
static constexpr int D_ = 128;
static constexpr float BETA_ = 0.25f;
static constexpr int NCHUNKS = 8;            // N split across blockIdx.y
static constexpr int NPER = N_ / NCHUNKS;    // 1024 codes per chunk

// ---------------------------------------------------------------------------
// helpers
// ---------------------------------------------------------------------------
__device__ __forceinline__ unsigned long long shfl_xor_u64(unsigned long long v, int mask) {
    unsigned int lo = (unsigned int)v;
    unsigned int hi = (unsigned int)(v >> 32);
    lo = __shfl_xor(lo, mask, 32);
    hi = __shfl_xor(hi, mask, 32);
    return ((unsigned long long)hi << 32) | (unsigned long long)lo;
}

// monotonic float -> uint mapping (order-preserving, incl. negatives)
__device__ __forceinline__ unsigned int sortable_f32(float f) {
    unsigned int u = __float_as_uint(f);
    return (u & 0x80000000u) ? ~u : (u | 0x80000000u);
}

// ---------------------------------------------------------------------------
// ||W_n||^2 for every codeword
// ---------------------------------------------------------------------------
__global__ __launch_bounds__(128) void vq_wn2_kernel(const float* __restrict__ W,
                                                     float* __restrict__ wn2) {
    int n = blockIdx.x * blockDim.x + threadIdx.x;
    if (n >= N_) return;
    const v4f* r = (const v4f*)(W + (size_t)n * D_);
    float s = 0.f;
#pragma unroll
    for (int i = 0; i < D_ / 4; ++i) {
        v4f v = r[i];
        s = fmaf(v.x, v.x, s); s = fmaf(v.y, v.y, s);
        s = fmaf(v.z, v.z, s); s = fmaf(v.w, v.w, s);
    }
    wn2[n] = s;
}

// ---------------------------------------------------------------------------
// x statistics: column sums sx[d] and total sum of squares
// ---------------------------------------------------------------------------
__global__ __launch_bounds__(128) void vq_xstats_kernel(const float* __restrict__ x,
                                                        float* __restrict__ colsum_x,
                                                        float* __restrict__ sum_x2) {
    __shared__ float red[128];
    int t = threadIdx.x;            // column d
    int r0 = blockIdx.x * 8;        // 8 rows per block
    float cs = 0.f, s2 = 0.f;
#pragma unroll
    for (int r = 0; r < 8; ++r) {
        float v = x[(size_t)(r0 + r) * D_ + t];
        cs += v;
        s2 = fmaf(v, v, s2);
    }
    atomicAdd(&colsum_x[t], cs);
    red[t] = s2;
    __syncthreads();
    for (int s = 64; s > 0; s >>= 1) {
        if (t < s) red[t] += red[t + s];
        __syncthreads();
    }
    if (t == 0) atomicAdd(sum_x2, red[0]);
}

// ---------------------------------------------------------------------------
// WMMA argmin: one wave per 16-row x tile x one N-chunk.
// score(b,n) = ||W_n||^2 - 2 * (x_b . W_n); argmin over n via packed u64 keys.
// ---------------------------------------------------------------------------
__global__ __launch_bounds__(32)
void vq_argmin_wmma(const float* __restrict__ x, const float* __restrict__ W,
                    const float* __restrict__ wn2,
                    unsigned long long* __restrict__ packed) {
    const int lane = threadIdx.x;   // 0..31, wave32
    const int half = lane >> 4;     // 0: K pair {0,1}; 1: K pair {2,3}
    const int l16  = lane & 15;
    const int row0 = blockIdx.x * 16;         // x rows of this tile
    const int ncol0 = blockIdx.y * NPER;      // codeword range

    // Preload A tile (16x128): lane holds row l16, K = 4s + 2*half + {0,1}
    v2f a[32];
    const float* xrow = x + (size_t)(row0 + l16) * D_ + 2 * half;
#pragma unroll
    for (int s = 0; s < 32; ++s) a[s] = *(const v2f*)(xrow + 4 * s);

    unsigned long long best[8];
#pragma unroll
    for (int v = 0; v < 8; ++v) best[v] = ~0ull;

    for (int t = 0; t < NPER / 16; ++t) {
        const int mycol = ncol0 + t * 16 + l16;       // B-matrix col = codeword id
        const float* wrow = W + (size_t)mycol * D_ + 2 * half;
        v8f c = {};
#pragma unroll
        for (int s = 0; s < 32; ++s) {
            v2f b = *(const v2f*)(wrow + 4 * s);      // B[k][n] = W[n][k]
            c = __builtin_amdgcn_wmma_f32_16x16x4_f32(
                    /*neg_a=*/false, a[s], /*neg_b=*/false, b,
                    /*c_mod=*/(short)0, c, /*reuse_a=*/false, /*reuse_b=*/false);
        }
        const float w2 = wn2[mycol];
#pragma unroll
        for (int v = 0; v < 8; ++v) {
            float score = w2 - 2.0f * c[v];           // row = v + 8*half, col = mycol
            unsigned long long key =
                ((unsigned long long)sortable_f32(score) << 32) | (unsigned int)mycol;
            best[v] = key < best[v] ? key : best[v];
        }
    }

    // reduce across the 16 lanes of each half (rows differ between halves)
#pragma unroll
    for (int v = 0; v < 8; ++v) {
        unsigned long long k = best[v];
#pragma unroll
        for (int off = 8; off >= 1; off >>= 1) {
            unsigned long long o = shfl_xor_u64(k, off);
            k = o < k ? o : k;
        }
        if (l16 == 0) atomicMin(&packed[row0 + v + 8 * half], k);
    }
}

// ---------------------------------------------------------------------------
// extract ind, gather q = W[ind], accumulate column sums sq[d] and sum q^2
// ---------------------------------------------------------------------------
__global__ __launch_bounds__(128)
void vq_qstats_kernel(const unsigned long long* __restrict__ packed,
                      const float* __restrict__ W,
                      float* __restrict__ colsum_q, float* __restrict__ sum_q2,
                      int* __restrict__ indbuf, float* __restrict__ out_ind) {
    __shared__ float red[128];
    int i = blockIdx.x;
    int t = threadIdx.x;
    int ind = (int)(packed[i] & 0xffffffffull);
    float v = W[(size_t)ind * D_ + t];
    atomicAdd(&colsum_q[t], v);
    red[t] = v * v;
    __syncthreads();
    for (int s = 64; s > 0; s >>= 1) {
        if (t < s) red[t] += red[t + s];
        __syncthreads();
    }
    if (t == 0) {
        atomicAdd(sum_q2, red[0]);
        indbuf[i] = ind;
        out_ind[i] = (float)ind;
    }
}

// ---------------------------------------------------------------------------
// loss = (1+beta) * (B*Sx2 - 2*sum_d sx[d]*sq[d] + B*Sq2) / (B*B*D)
// ---------------------------------------------------------------------------
__global__ __launch_bounds__(128)
void vq_loss_kernel(const float* __restrict__ colsum_x, const float* __restrict__ colsum_q,
                    const float* __restrict__ sum_x2, const float* __restrict__ sum_q2,
                    float* __restrict__ out_loss) {
    __shared__ float red[128];
    int t = threadIdx.x;
    red[t] = colsum_x[t] * colsum_q[t];
    __syncthreads();
    for (int s = 64; s > 0; s >>= 1) {
        if (t < s) red[t] += red[t + s];
        __syncthreads();
    }
    if (t == 0) {
        float total = (float)B_ * (*sum_x2) - 2.0f * red[0] + (float)B_ * (*sum_q2);
        *out_loss = (1.0f + BETA_) * total / ((float)B_ * (float)B_ * (float)D_);
    }
}

// ---------------------------------------------------------------------------
// quantized[i,j,d] = W[ind[i], d]  -- 537 MB streaming store (NT hint)
// ---------------------------------------------------------------------------
__global__ __launch_bounds__(256)
void vq_bcast_kernel(const int* __restrict__ indbuf, const float* __restrict__ W,
                     float* __restrict__ out) {
    int i = blockIdx.x;
    int ind = __ldg(&indbuf[i]);
    int vec = threadIdx.x & 31;     // which float4 of the 128-float row
    int j0  = threadIdx.x >> 5;     // 0..7
    v4f q = ((const v4f*)(W + (size_t)ind * D_))[vec];
    float* base = out + (size_t)i * B_ * D_;
    for (int j = j0; j < B_; j += 8) {
        __builtin_nontemporal_store(q, (v4f*)(base + (size_t)j * D_) + vec);
    }
}

// ---------------------------------------------------------------------------
extern "C" void kernel_launch(void* const* d_in, const int* in_sizes, int n_in,
                              void* d_out, int out_size, void* d_ws, size_t ws_size,
                              hipStream_t stream) {
    const float* x = (const float*)d_in[0];   // (B, D)
    const float* W = (const float*)d_in[1];   // (N, D)
    float* out = (float*)d_out;               // [quantized (B*B*D) | ind (B) | loss (1)]

    // workspace layout
    unsigned long long* packed = (unsigned long long*)d_ws;            //  8 KB
    float* wn2      = (float*)((char*)d_ws + 8192);                    // 32 KB
    float* colsum_x = (float*)((char*)d_ws + 8192 + 32768);            // 512 B
    float* colsum_q = colsum_x + 128;                                  // 512 B
    float* sums     = colsum_q + 128;                                  // sums[0]=Sx2, sums[1]=Sq2
    int*   indbuf   = (int*)(sums + 2);                                //  4 KB

    float* out_ind  = out + (size_t)B_ * B_ * D_;
    float* out_loss = out_ind + B_;

    hipMemsetAsync(packed, 0xFF, (size_t)B_ * sizeof(unsigned long long), stream);
    hipMemsetAsync(colsum_x, 0, (128 + 128 + 2) * sizeof(float), stream);

    vq_wn2_kernel<<<N_ / 128, 128, 0, stream>>>(W, wn2);
    vq_xstats_kernel<<<B_ / 8, 128, 0, stream>>>(x, colsum_x, sums);
    vq_argmin_wmma<<<dim3(B_ / 16, NCHUNKS), 32, 0, stream>>>(x, W, wn2, packed);
    vq_qstats_kernel<<<B_, 128, 0, stream>>>(packed, W, colsum_q, sums + 1, indbuf, out_ind);
    vq_loss_kernel<<<1, 128, 0, stream>>>(colsum_x, colsum_q, sums, sums + 1, out_loss);
    vq_bcast_kernel<<<B_, 256, 0, stream>>>(indbuf, W, out);
}